// MemNN_65781719105902
// MI455X (gfx1250) — compile-verified
//
#include <hip/hip_runtime.h>
#include <hip/hip_bf16.h>
#include <math.h>

// Problem constants (from reference)
#define VOCAB 50000
#define EMBD  128
#define HOPS  3
#define BS    64
#define STORY 50
#define SENT  20
#define QLEN  20

typedef __attribute__((ext_vector_type(2))) float v2f;
typedef __attribute__((ext_vector_type(8))) float v8f;

// ---------------------------------------------------------------------------
// Kernel 1: u0[b][d] = sum_j A[0][q[b][j]][d]
// ---------------------------------------------------------------------------
__global__ __launch_bounds__(EMBD) void memnn_u0(const int* __restrict__ q,
                                                 const float* __restrict__ A,
                                                 float* __restrict__ u) {
    __shared__ int tok[QLEN];
    const int b = blockIdx.x;
    const int d = threadIdx.x;
    if (d < QLEN) tok[d] = q[b * QLEN + d];
    __syncthreads();
    float acc = 0.0f;
#pragma unroll
    for (int j = 0; j < QLEN; ++j)
        acc += A[(size_t)tok[j] * EMBD + d];
    u[b * EMBD + d] = acc;
}

// ---------------------------------------------------------------------------
// Kernel 2: fused gather. For each (b,s) read the 20 token rows of all 4
// embedding levels ONCE and produce:
//   m[h][b][s][d] = sum_j A[h][x]*pe(j,d)   + TA[s][d]   (h = 0,1,2)
//   c[h][b][s][d] = sum_j A[h+1][x]          + TC[s][d]   (h = 0,1,2)
// This is the data-movement-optimal form: 4 table gathers instead of 6.
// ---------------------------------------------------------------------------
__global__ __launch_bounds__(EMBD) void memnn_gather(const int* __restrict__ x,
                                                     const float* __restrict__ A,
                                                     const float* __restrict__ TA,
                                                     const float* __restrict__ TC,
                                                     float* __restrict__ m,
                                                     float* __restrict__ c) {
    const int bs = blockIdx.x;          // b*STORY + s
    const int s  = bs % STORY;
    const int d  = threadIdx.x;
    __shared__ int tok[SENT];
    if (d < SENT) tok[d] = x[(size_t)bs * SENT + d];
    __syncthreads();

    const float kd = (float)(d + 1) / (float)EMBD;
    const size_t LVL = (size_t)VOCAB * EMBD;

    float mp0 = 0.f, mp1 = 0.f, mp2 = 0.f;
    float cp1 = 0.f, cp2 = 0.f, cp3 = 0.f;
#pragma unroll
    for (int j = 0; j < SENT; ++j) {
        const float jj = (float)(j + 1) / (float)SENT;
        const float pe = 1.0f - jj - kd * (1.0f - 2.0f * jj);
        const size_t base = (size_t)tok[j] * EMBD + d;
        const float e0 = A[base];
        const float e1 = A[LVL + base];
        const float e2 = A[2 * LVL + base];
        const float e3 = A[3 * LVL + base];
        mp0 += e0 * pe;  mp1 += e1 * pe;  mp2 += e2 * pe;
        cp1 += e1;       cp2 += e2;       cp3 += e3;
    }
    const float ta = TA[s * EMBD + d];
    const float tc = TC[s * EMBD + d];
    const size_t off = (size_t)bs * EMBD + d;
    const size_t hs  = (size_t)BS * STORY * EMBD;   // per-hop stride
    m[off]          = mp0 + ta;
    m[hs + off]     = mp1 + ta;
    m[2 * hs + off] = mp2 + ta;
    c[off]          = cp1 + tc;
    c[hs + off]     = cp2 + tc;
    c[2 * hs + off] = cp3 + tc;
}

// ---------------------------------------------------------------------------
// Kernel 3: one hop. 64 blocks (batch), 128 threads (=4 wave32).
//   score[s] = m[h][b][s] . u[b]     (wave-parallel dot + shfl reduce)
//   p = softmax(score) ; u[b] += p @ c[h][b]
// ---------------------------------------------------------------------------
__global__ __launch_bounds__(EMBD) void memnn_hop(const float* __restrict__ m,
                                                  const float* __restrict__ c,
                                                  float* __restrict__ u,
                                                  int hop) {
    const int b = blockIdx.x;
    const int t = threadIdx.x;
    __shared__ float u_lds[EMBD];
    __shared__ float score[STORY];
    u_lds[t] = u[b * EMBD + t];
    __syncthreads();

    const int wave = t >> 5;
    const int lane = t & 31;
    const size_t hb = (size_t)hop * BS * STORY * EMBD + (size_t)b * STORY * EMBD;

    for (int s = wave; s < STORY; s += 4) {
        const float* mrow = m + hb + (size_t)s * EMBD;
        float p = 0.f;
#pragma unroll
        for (int i = 0; i < EMBD / 32; ++i)
            p += mrow[lane + 32 * i] * u_lds[lane + 32 * i];
        for (int o = 16; o > 0; o >>= 1) p += __shfl_xor(p, o, 32);
        if (lane == 0) score[s] = p;
    }
    __syncthreads();

    float mx = -3.4e38f;
    for (int s = 0; s < STORY; ++s) mx = fmaxf(mx, score[s]);
    float sm = 0.f;
    for (int s = 0; s < STORY; ++s) sm += __expf(score[s] - mx);
    const float inv = 1.0f / sm;

    float acc = 0.f;
    const float* crow = c + hb;
    for (int s = 0; s < STORY; ++s)
        acc += (__expf(score[s] - mx) * inv) * crow[(size_t)s * EMBD + t];
    u[b * EMBD + t] = u_lds[t] + acc;
}

// ---------------------------------------------------------------------------
// Kernel 4: logits = u[64,128] @ A3[50000,128]^T via V_WMMA_F32_16X16X4_F32.
// One 16x16 output tile per wave; 32 WMMAs over K=128. u is staged in LDS
// (32 KB) once per block. 12500 tiles total (4 M-tiles x 3125 N-tiles).
// Layouts per CDNA5 ISA §7.12.2:
//   A 16x4 f32 : lane l holds A[l%16][kk*4 + v + 2*(l/16)], v=0..1
//   B  4x16 f32: lane l holds B[kk*4 + v + 2*(l/16)][l%16]
//   D 16x16 f32: lane l holds D[v + 8*(l/16)][l%16], v=0..7
// ---------------------------------------------------------------------------
__global__ __launch_bounds__(256) void memnn_logits_wmma(const float* __restrict__ u,
                                                         const float* __restrict__ A3,
                                                         float* __restrict__ out) {
    __shared__ float u_lds[BS * EMBD];        // 32 KB
    for (int i = threadIdx.x; i < BS * EMBD; i += 256) u_lds[i] = u[i];
    __syncthreads();

    const int wave = threadIdx.x >> 5;        // 8 waves / block
    const int lane = threadIdx.x & 31;
    const int tile = blockIdx.x * 8 + wave;   // wave-uniform
    const int NTILES = (VOCAB / 16) * (BS / 16);  // 12500
    if (tile >= NTILES) return;               // uniform exit: EXEC all-1 for WMMA

    const int mb = (tile & 3) << 4;           // batch tile base
    const int nb = (tile >> 2) << 4;          // vocab tile base
    const int n  = lane & 15;
    const int h  = lane >> 4;

    const float* arow = u_lds + (mb + n) * EMBD;
    const float* brow = A3 + (size_t)(nb + n) * EMBD;

    v8f acc = {};
#pragma unroll
    for (int kk = 0; kk < EMBD / 4; ++kk) {
        const int k0 = kk * 4 + 2 * h;
        v2f a;  a.x  = arow[k0]; a.y  = arow[k0 + 1];
        v2f bv; bv.x = brow[k0]; bv.y = brow[k0 + 1];
        acc = __builtin_amdgcn_wmma_f32_16x16x4_f32(
            /*neg_a=*/false, a, /*neg_b=*/false, bv,
            /*c_mod=*/(short)0, acc, /*reuse_a=*/false, /*reuse_b=*/false);
    }
#pragma unroll
    for (int v = 0; v < 8; ++v)
        out[(size_t)(mb + v + 8 * h) * VOCAB + (nb + n)] = acc[v];
}

// ---------------------------------------------------------------------------
// Kernel 5: row-wise log_softmax over 50000, in place in d_out.
// ---------------------------------------------------------------------------
__global__ __launch_bounds__(256) void memnn_logsoftmax(float* __restrict__ out) {
    __shared__ float red[256];
    const int b = blockIdx.x;
    const int t = threadIdx.x;
    float* row = out + (size_t)b * VOCAB;

    float mx = -3.4e38f;
    for (int i = t; i < VOCAB; i += 256) mx = fmaxf(mx, row[i]);
    red[t] = mx; __syncthreads();
    for (int o = 128; o > 0; o >>= 1) {
        if (t < o) red[t] = fmaxf(red[t], red[t + o]);
        __syncthreads();
    }
    mx = red[0]; __syncthreads();

    float sm = 0.f;
    for (int i = t; i < VOCAB; i += 256) sm += __expf(row[i] - mx);
    red[t] = sm; __syncthreads();
    for (int o = 128; o > 0; o >>= 1) {
        if (t < o) red[t] += red[t + o];
        __syncthreads();
    }
    const float lse = mx + logf(red[0]);

    for (int i = t; i < VOCAB; i += 256) row[i] = row[i] - lse;
}

// ---------------------------------------------------------------------------
extern "C" void kernel_launch(void* const* d_in, const int* in_sizes, int n_in,
                              void* d_out, int out_size, void* d_ws, size_t ws_size,
                              hipStream_t stream) {
    const int*   x  = (const int*)d_in[0];    // [64,50,20]
    const int*   q  = (const int*)d_in[1];    // [64,20]
    const float* A  = (const float*)d_in[2];  // [4,50000,128]
    const float* TA = (const float*)d_in[3];  // [1,50,128]
    const float* TC = (const float*)d_in[4];  // [1,50,128]
    float* out = (float*)d_out;               // [64,50000]

    float* ws = (float*)d_ws;
    float* u  = ws;                                   // 64*128
    float* m  = ws + BS * EMBD;                       // 3*64*50*128
    float* c  = m + (size_t)HOPS * BS * STORY * EMBD; // 3*64*50*128

    memnn_u0<<<BS, EMBD, 0, stream>>>(q, A, u);
    memnn_gather<<<BS * STORY, EMBD, 0, stream>>>(x, A, TA, TC, m, c);
    for (int h = 0; h < HOPS; ++h)
        memnn_hop<<<BS, EMBD, 0, stream>>>(m, c, u, h);

    const int ntiles = (VOCAB / 16) * (BS / 16);      // 12500
    const int blocks = (ntiles + 7) / 8;              // 8 waves/block
    memnn_logits_wmma<<<blocks, 256, 0, stream>>>(
        u, A + 3 * (size_t)VOCAB * EMBD, out);

    memnn_logsoftmax<<<BS, 256, 0, stream>>>(out);
}